// CPCLossNCE_2834678415380
// MI455X (gfx1250) — compile-verified
//
#include <hip/hip_runtime.h>

typedef __attribute__((ext_vector_type(2))) float v2f;
typedef __attribute__((ext_vector_type(8))) float v8f;
typedef __attribute__((ext_vector_type(4))) int   v4i;

#define B_     64
#define D_     512
#define H_     8
#define W_     8
#define P_     28          // triu pairs of H
#define NB_    64          // H*W
#define NTOT_  4096        // B*NB
#define MROWS_ 14336       // P*W*B

#define NCE_ROWS_PER_BLOCK 64
#define NCE_BLOCKS (MROWS_ / NCE_ROWS_PER_BLOCK)   // 224

// CDNA5 async global->LDS staging (builtin exists on this toolchain; its
// params are AS1/AS3 int4 pointers per the round-2 diagnostic)
#if defined(__has_builtin)
#if __has_builtin(__builtin_amdgcn_global_load_async_to_lds_b128) && \
    __has_builtin(__builtin_amdgcn_s_wait_asynccnt)
#define HAVE_ASYNC_LDS 1
typedef v4i __attribute__((address_space(1)))* as1_v4i_ptr;
typedef v4i __attribute__((address_space(3)))* as3_v4i_ptr;
#endif
#endif

// decode p -> (i,k) of np.triu_indices(8, k=1) (row-major over i)
__device__ __forceinline__ void decode_pair(int p, int& i_out, int& k_out) {
    int acc = 0;
    for (int i = 0; i < H_ - 1; ++i) {
        int cnt = H_ - 1 - i;
        if (p < acc + cnt) { i_out = i; k_out = i + 1 + (p - acc); return; }
        acc += cnt;
    }
    i_out = 0; k_out = 1;
}

// fp32 WMMA: D(16x16) = A(16x4) * B(4x16) + C, full fp32 precision.
__device__ __forceinline__ v8f wmma_f32(v2f a, v2f b, v8f c) {
    return __builtin_amdgcn_wmma_f32_16x16x4_f32(
        /*neg_a=*/false, a, /*neg_b=*/false, b,
        /*c_mod=*/(short)0, c, /*reuse_a=*/false, /*reuse_b=*/false);
}

// ---------------------------------------------------------------------------
// Kernel 1: zh[p, m=(w,b), e] = sum_d C[b,d,ii[p],w] * Wk[kk[p]-1, e, d] + bk
// 28 GEMMs of 512x512x512. grid = P*4*2; block = 256 (8 waves); each wave
// computes a 64x64 tile as 4x4 WMMA accumulators.
// ---------------------------------------------------------------------------
__global__ __launch_bounds__(256) void zh_gemm_kernel(
    const float* __restrict__ Cin, const float* __restrict__ Wk,
    const float* __restrict__ bk, float* __restrict__ zh)
{
    int gid  = blockIdx.x;
    int p    = gid >> 3;
    int mblk = (gid >> 1) & 3;   // 4 blocks of 128 rows
    int nblk = gid & 1;          // 2 blocks of 256 cols
    int ip, kp; decode_pair(p, ip, kp);
    const float* Wg = Wk + (size_t)(kp - 1) * D_ * D_;   // Wg[e*D_ + d]
    const float* bg = bk + (size_t)(kp - 1) * D_;

    int lane = threadIdx.x & 31;
    int wave = threadIdx.x >> 5;          // 0..7 -> 2(M) x 4(N)
    int wm = wave & 1, wn = wave >> 1;
    int M0 = mblk * 128 + wm * 64;
    int N0 = nblk * 256 + wn * 64;

    int lr = lane & 15;                   // M/N index within fragment
    int kh = lane >> 4;                   // 0/1 -> K offset 0/2

    v8f acc[4][4];
    for (int mt = 0; mt < 4; ++mt)
        for (int nt = 0; nt < 4; ++nt)
            for (int r = 0; r < 8; ++r) acc[mt][nt][r] = 0.0f;

    // per-lane A base: A[m][d] = C[b, d, ii, w] ; m=(w<<6)|b
    int abase[4];
    for (int mt = 0; mt < 4; ++mt) {
        int m = M0 + mt * 16 + lr;
        int widx = m >> 6, bidx = m & 63;
        abase[mt] = bidx * (D_ * NB_) + ip * W_ + widx;
    }
    int bcol[4];
    for (int nt = 0; nt < 4; ++nt) bcol[nt] = (N0 + nt * 16 + lr) * D_;

    for (int d0 = 0; d0 < D_; d0 += 4) {
        int k0 = d0 + kh * 2;
        v2f a[4], bb[4];
        for (int mt = 0; mt < 4; ++mt) {
            a[mt].x = Cin[abase[mt] + (size_t)k0 * NB_];
            a[mt].y = Cin[abase[mt] + (size_t)(k0 + 1) * NB_];
        }
        for (int nt = 0; nt < 4; ++nt)
            bb[nt] = *(const v2f*)(Wg + bcol[nt] + k0);   // contiguous in d
        for (int mt = 0; mt < 4; ++mt)
            for (int nt = 0; nt < 4; ++nt)
                acc[mt][nt] = wmma_f32(a[mt], bb[nt], acc[mt][nt]);
    }

    // epilogue: bias + store (D layout: row = kh*8+r, col = lr)
    for (int nt = 0; nt < 4; ++nt) {
        float bias = bg[N0 + nt * 16 + lr];
        for (int mt = 0; mt < 4; ++mt) {
            int mbase = M0 + mt * 16 + kh * 8;
            float* dst = zh + ((size_t)p * 512 + mbase) * D_ + N0 + nt * 16 + lr;
            for (int r = 0; r < 8; ++r)
                dst[(size_t)r * D_] = acc[mt][nt][r] + bias;
        }
    }
}

// ---------------------------------------------------------------------------
// Kernel 2: pos[row] = sum_e zh[row,e] * Z[b, e, kk[p], w]  (one wave / row)
// ---------------------------------------------------------------------------
__global__ __launch_bounds__(128) void pos_kernel(
    const float* __restrict__ Z, const float* __restrict__ zh,
    float* __restrict__ pos)
{
    int row  = blockIdx.x * 4 + (threadIdx.x >> 5);
    int lane = threadIdx.x & 31;
    int p = row >> 9;
    int m = row & 511;
    int widx = m >> 6, bidx = m & 63;
    int ip, kp; decode_pair(p, ip, kp); (void)ip;
    const float* zr = zh + (size_t)row * D_;
    const float* Zr = Z + (size_t)bidx * (D_ * NB_) + kp * W_ + widx;
    float s = 0.0f;
    for (int e = lane; e < D_; e += 32)
        s += zr[e] * Zr[(size_t)e * NB_];
    for (int off = 16; off > 0; off >>= 1)
        s += __shfl_xor(s, off, 32);
    if (lane == 0) pos[row] = s;
}

// ---------------------------------------------------------------------------
// Kernel 3: raw = zh @ Zneg fused with masked online logsumexp over N=4096.
// Block = 64 rows (exactly one (p,w): rows are b=0..63) x all N. 8 waves as
// 4(M, 16 rows) x 2(N, 2048 cols = 32 groups of 64; each group = one b').
// A panel (64x512) staged via async global->LDS into dynamic LDS (132KB of
// the 320KB WGP pool) with pad-516 stride -> conflict-free ds_load_b64
// fragment reads. B streamed from global (L2 resident; next slab prefetched
// via global_prefetch_b8).
// ---------------------------------------------------------------------------
#define LDSA_STRIDE 516
#define NCE_LDS_BYTES (NCE_ROWS_PER_BLOCK * LDSA_STRIDE * 4)

__global__ __launch_bounds__(256) void nce_kernel(
    const float* __restrict__ Z, const float* __restrict__ zh,
    const float* __restrict__ pos, float* __restrict__ blocksums)
{
    extern __shared__ float ldsA[];                 // [64][LDSA_STRIDE]
    __shared__ float stat_m[2][NCE_ROWS_PER_BLOCK];
    __shared__ float stat_s[2][NCE_ROWS_PER_BLOCK];
    __shared__ float contrib[NCE_ROWS_PER_BLOCK];

    int rb = blockIdx.x;                            // 0..223
    int rowbase = rb * NCE_ROWS_PER_BLOCK;          // one (p,w); rows are b
    int tid  = threadIdx.x;
    int lane = tid & 31;
    int wave = tid >> 5;                            // 0..7
    int wm = wave & 3;                              // M sub-block (16 rows)
    int wn = wave >> 2;                             // N half (2048 cols)
    int lr = lane & 15, kh = lane >> 4;

    // ---- stage A (64 x 512) into LDS ----
#if defined(HAVE_ASYNC_LDS)
    for (int i = tid; i < NCE_ROWS_PER_BLOCK * (D_ / 4); i += 256) {
        int r  = i >> 7;                            // 128 float4 per row
        int c4 = i & 127;
        const float* g = zh + (size_t)(rowbase + r) * D_ + c4 * 4;
        float* l = &ldsA[r * LDSA_STRIDE + c4 * 4];
        __builtin_amdgcn_global_load_async_to_lds_b128(
            (as1_v4i_ptr)(v4i*)g, (as3_v4i_ptr)(v4i*)l, 0, 0);
    }
    __builtin_amdgcn_s_wait_asynccnt(0);
#else
    {
        const float4* src = (const float4*)(zh + (size_t)rowbase * D_);
        for (int i = tid; i < NCE_ROWS_PER_BLOCK * (D_ / 4); i += 256) {
            int r  = i >> 7;
            int c4 = i & 127;
            float4 v = src[r * (D_ / 4) + c4];
            float* dst = &ldsA[r * LDSA_STRIDE + c4 * 4];
            dst[0] = v.x; dst[1] = v.y; dst[2] = v.z; dst[3] = v.w;
        }
    }
#endif
    __syncthreads();

    float m_run[8], s_run[8];
    for (int r = 0; r < 8; ++r) { m_run[r] = -3.0e38f; s_run[r] = 0.0f; }

    int arow = (wm * 16 + lr) * LDSA_STRIDE;        // this lane's A panel row

    for (int gi = 0; gi < 32; ++gi) {
        int bprime = wn * 32 + gi;                  // negative batch index
        const float* Zb = Z + (size_t)bprime * (D_ * NB_);

        // prefetch next negative slab (emits global_prefetch_b8)
        if (gi + 1 < 32)
            __builtin_prefetch(Zb + (size_t)(D_ * NB_) + lane * 64, 0, 0);

        v8f acc[4];
        for (int nt = 0; nt < 4; ++nt)
            for (int r = 0; r < 8; ++r) acc[nt][r] = 0.0f;

        for (int d0 = 0; d0 < D_; d0 += 4) {
            int k0 = d0 + kh * 2;
            v2f a = *(const v2f*)&ldsA[arow + k0];
            for (int nt = 0; nt < 4; ++nt) {
                int j = nt * 16 + lr;               // col within group
                v2f bfrag;
                bfrag.x = Zb[(size_t)k0 * NB_ + j];
                bfrag.y = Zb[(size_t)(k0 + 1) * NB_ + j];
                acc[nt] = wmma_f32(a, bfrag, acc[nt]);
            }
        }
        // online (max, sumexp) update; mask excludes one whole group per row
        for (int r = 0; r < 8; ++r) {
            int brow = wm * 16 + kh * 8 + r;        // == b for this row
            if (brow == bprime) continue;           // masked (diagonal) group
            float x0 = acc[0][r], x1 = acc[1][r], x2 = acc[2][r], x3 = acc[3][r];
            float mx = fmaxf(fmaxf(x0, x1), fmaxf(x2, x3));
            float mnew = fmaxf(m_run[r], mx);
            s_run[r] = s_run[r] * __expf(m_run[r] - mnew)
                     + __expf(x0 - mnew) + __expf(x1 - mnew)
                     + __expf(x2 - mnew) + __expf(x3 - mnew);
            m_run[r] = mnew;
        }
    }

    // reduce (m,s) across the 16 lanes of each half-wave
    for (int off = 1; off < 16; off <<= 1) {
        for (int r = 0; r < 8; ++r) {
            float om = __shfl_xor(m_run[r], off, 32);
            float os = __shfl_xor(s_run[r], off, 32);
            float mnew = fmaxf(m_run[r], om);
            s_run[r] = s_run[r] * __expf(m_run[r] - mnew) + os * __expf(om - mnew);
            m_run[r] = mnew;
        }
    }
    if (lr == 0) {
        for (int r = 0; r < 8; ++r) {
            stat_m[wn][wm * 16 + kh * 8 + r] = m_run[r];
            stat_s[wn][wm * 16 + kh * 8 + r] = s_run[r];
        }
    }
    __syncthreads();

    if (tid < NCE_ROWS_PER_BLOCK) {
        float m = stat_m[0][tid], s = stat_s[0][tid];
        {
            float om = stat_m[1][tid], os = stat_s[1][tid];
            float mnew = fmaxf(m, om);
            s = s * __expf(m - mnew) + os * __expf(om - mnew);
            m = mnew;
        }
        // reference quirk: masked entries are literal zeros inside the max
        float mf = fmaxf(m, 0.0f);
        s = s * __expf(m - mf);
        float ps = pos[rowbase + tid] - mf;
        contrib[tid] = __logf(__expf(ps) + s) - ps;   // = -nce(row)
    }
    __syncthreads();
    if (tid == 0) {
        float a = 0.0f;
        for (int r = 0; r < NCE_ROWS_PER_BLOCK; ++r) a += contrib[r];
        blocksums[rb] = a;
    }
}

// ---------------------------------------------------------------------------
// Kernel 4: deterministic final reduction -> out[0] = mean over 14336 rows
// ---------------------------------------------------------------------------
__global__ __launch_bounds__(256) void finalize_kernel(
    const float* __restrict__ bs, float* __restrict__ out)
{
    __shared__ float sh[256];
    float s = 0.0f;
    for (int i = threadIdx.x; i < NCE_BLOCKS; i += 256) s += bs[i];
    sh[threadIdx.x] = s;
    __syncthreads();
    for (int off = 128; off > 0; off >>= 1) {
        if (threadIdx.x < off) sh[threadIdx.x] += sh[threadIdx.x + off];
        __syncthreads();
    }
    if (threadIdx.x == 0) out[0] = sh[0] / (float)MROWS_;
}

// ---------------------------------------------------------------------------
extern "C" void kernel_launch(void* const* d_in, const int* in_sizes, int n_in,
                              void* d_out, int out_size, void* d_ws, size_t ws_size,
                              hipStream_t stream)
{
    (void)in_sizes; (void)n_in; (void)out_size; (void)ws_size;
    const float* Z   = (const float*)d_in[0];   // (64,512,8,8)
    const float* Cin = (const float*)d_in[1];   // (64,512,8,8)
    const float* Wk  = (const float*)d_in[2];   // (7,512,512)
    const float* bk  = (const float*)d_in[3];   // (7,512)
    float* out = (float*)d_out;

    float* ws  = (float*)d_ws;
    float* zh        = ws;                       // 7,340,032 floats (29.4 MB)
    float* pos       = ws + 7340032;             // 14,336 floats
    float* blocksums = ws + 7354368;             // 224 floats

    zh_gemm_kernel<<<P_ * 4 * 2, 256, 0, stream>>>(Cin, Wk, bk, zh);
    pos_kernel<<<MROWS_ / 4, 128, 0, stream>>>(Z, zh, pos);
    nce_kernel<<<NCE_BLOCKS, 256, NCE_LDS_BYTES, stream>>>(Z, zh, pos, blocksums);
    finalize_kernel<<<1, 256, 0, stream>>>(blocksums, out);
}